// GCNEncoder_53120155517005
// MI455X (gfx1250) — compile-verified
//
#include <hip/hip_runtime.h>
#include <hip/hip_bf16.h>

typedef float v2f __attribute__((ext_vector_type(2)));
typedef float v8f __attribute__((ext_vector_type(8)));

// ---------------------------------------------------------------------------
// Elementwise fill (used to zero accumulators / init degree with self-loop 1)
// ---------------------------------------------------------------------------
__global__ void k_fill(float* __restrict__ p, float val, long long n) {
    long long i = (long long)blockIdx.x * blockDim.x + threadIdx.x;
    if (i < n) p[i] = val;
}

// ---------------------------------------------------------------------------
// Degree accumulation over edge destinations (float atomics; counts are exact)
// ---------------------------------------------------------------------------
__global__ void k_deg_edges(const long long* __restrict__ dst,
                            float* __restrict__ deg, long long E) {
    long long e = (long long)blockIdx.x * blockDim.x + threadIdx.x;
    if (e < E) atomicAdd(&deg[dst[e]], 1.0f);
}

__global__ void k_rsqrt_inplace(float* __restrict__ d, int N) {
    int i = blockIdx.x * blockDim.x + threadIdx.x;
    if (i < N) {
        float v = d[i];
        d[i] = (v > 0.0f) ? rsqrtf(v) : 0.0f;
    }
}

// ---------------------------------------------------------------------------
// H = X @ W  via V_WMMA_F32_16X16X4_F32 (full fp32 precision).
// X: [N,64] row-major, W: [64,COUT] row-major, H: [N,COUT].
// One wave32 computes one 16x16 output tile; K=64 = 16 chained K=4 WMMAs.
// A layout (16x4 f32): lane = M (lane&15); VGPR0/1 = K = 2*(lane>>4) + {0,1}.
// B layout (4x16 f32): lane = N (lane&15); VGPR0/1 = K = 2*(lane>>4) + {0,1}.
// D layout (16x16 f32): VGPR j -> M = j + 8*(lane>>4), N = lane&15.
// Tile predicate is wave-uniform and loads are clamped, so EXEC is all-ones
// across the WMMA as the ISA requires.
// ---------------------------------------------------------------------------
template <int COUT>
__global__ void k_gemm_wmma(const float* __restrict__ X,
                            const float* __restrict__ W,
                            float* __restrict__ H, int N) {
    const int lane  = threadIdx.x & 31;
    const int gwave = blockIdx.x * (blockDim.x >> 5) + (threadIdx.x >> 5);
    const int tilesN = COUT >> 4;
    const int tm = gwave / tilesN;         // node-tile index (wave-uniform)
    const int tn = gwave - tm * tilesN;    // channel-tile index
    if (tm * 16 >= N) return;              // wave-uniform exit

    int row = tm * 16 + (lane & 15);
    if (row >= N) row = N - 1;             // clamp loads; stores guarded below
    const int col = tn * 16 + (lane & 15);
    const int kb  = (lane >> 4) << 1;      // 0 or 2

    const float* xrow = X + (long long)row * 64;
    v8f acc = {};
#pragma unroll
    for (int k = 0; k < 64; k += 4) {
        v2f a, b;
        a[0] = xrow[k + kb];
        a[1] = xrow[k + kb + 1];
        b[0] = W[(long long)(k + kb) * COUT + col];
        b[1] = W[(long long)(k + kb + 1) * COUT + col];
        acc = __builtin_amdgcn_wmma_f32_16x16x4_f32(
            /*neg_a=*/false, a, /*neg_b=*/false, b,
            /*c_mod=*/(short)0, acc, /*reuse_a=*/false, /*reuse_b=*/false);
    }

    const int m0 = tm * 16 + ((lane >> 4) << 3);
    const int n  = tn * 16 + (lane & 15);
#pragma unroll
    for (int j = 0; j < 8; ++j) {
        const int m = m0 + j;
        if (m < N) H[(long long)m * COUT + n] = acc[j];
    }
}

// ---------------------------------------------------------------------------
// Edge scatter: AGG[dst] += dinv[src]*dinv[dst] * H[src]  (no self-loops here;
// they are folded into the epilogue). C/4 lanes cooperate per edge: coalesced
// float4 gathers of the 256B/128B source row, 4 consecutive f32 atomics out.
// With the ~90MB working set resident in the 192MB L2, both the random gather
// and the atomics resolve in L2.
// ---------------------------------------------------------------------------
template <int C>
__global__ void k_scatter(const long long* __restrict__ src,
                          const long long* __restrict__ dst,
                          const float* __restrict__ dinv,
                          const float* __restrict__ H,
                          float* __restrict__ AGG, long long E) {
    constexpr int CH = C / 4;  // float4 chunks per edge (power of two)
    long long tid = (long long)blockIdx.x * blockDim.x + threadIdx.x;
    long long e  = tid / CH;
    int       ch = (int)(tid - e * CH);
    if (e >= E) return;

    const long long s = src[e];
    const long long d = dst[e];
    const float norm = dinv[s] * dinv[d];

    const float4 v = *(const float4*)(H + s * C + (ch << 2));
    float* o = AGG + d * C + (ch << 2);
    atomicAdd(o + 0, v.x * norm);
    atomicAdd(o + 1, v.y * norm);
    atomicAdd(o + 2, v.z * norm);
    atomicAdd(o + 3, v.w * norm);
}

// ---------------------------------------------------------------------------
// Epilogue: OUT = [relu]( AGG + dinv^2 * H + bias ), in place on AGG.
// The dinv^2*H term is the self-loop edge contribution.
// ---------------------------------------------------------------------------
template <int C, bool RELU>
__global__ void k_finish(float* __restrict__ OUT, const float* __restrict__ H,
                         const float* __restrict__ dinv,
                         const float* __restrict__ bias, int N) {
    long long idx = (long long)blockIdx.x * blockDim.x + threadIdx.x;
    if (idx >= (long long)N * C) return;
    const int i = (int)(idx / C);
    const int c = (int)(idx - (long long)i * C);
    const float di = dinv[i];
    float v = OUT[idx] + di * di * H[idx] + bias[c];
    OUT[idx] = RELU ? fmaxf(v, 0.0f) : v;
}

// ---------------------------------------------------------------------------
// Host-side orchestration
// ---------------------------------------------------------------------------
extern "C" void kernel_launch(void* const* d_in, const int* in_sizes, int n_in,
                              void* d_out, int out_size, void* d_ws, size_t ws_size,
                              hipStream_t stream) {
    (void)n_in; (void)out_size; (void)ws_size;

    const float*     x    = (const float*)d_in[0];
    const long long* eidx = (const long long*)d_in[1];  // int64 [2, E]
    // d_in[2] = num_nodes scalar (derived from sizes instead)
    const float* W1 = (const float*)d_in[3];
    const float* b1 = (const float*)d_in[4];
    const float* W2 = (const float*)d_in[5];
    const float* b2 = (const float*)d_in[6];

    const int       N = in_sizes[0] / 64;   // x is [N, 64]
    const long long E = (long long)in_sizes[1] / 2;
    const long long* src = eidx;
    const long long* dst = eidx + E;

    // Workspace layout (fp32): dinv[N] | h1[N*64] | z1[N*64] | h2[N*32]
    float* dinv = (float*)d_ws;
    float* h1   = dinv + N;
    float* z1   = h1 + (long long)N * 64;   // agg1, then ReLU output in place
    float* h2   = z1 + (long long)N * 64;
    float* out  = (float*)d_out;            // agg2 accumulates directly here

    const int B = 256;  // 8 waves per block on wave32
    auto cdiv = [](long long a, long long b) { return (int)((a + b - 1) / b); };

    // Symmetric-normalization coefficients: deg (self-loop=1) -> rsqrt in place
    k_fill<<<cdiv(N, B), B, 0, stream>>>(dinv, 1.0f, N);
    k_deg_edges<<<cdiv(E, B), B, 0, stream>>>(dst, dinv, E);
    k_rsqrt_inplace<<<cdiv(N, B), B, 0, stream>>>(dinv, N);

    // ---- Layer 1: h1 = x @ W1 ; agg ; relu(agg + dinv^2*h1 + b1) ----
    {
        const int waves = cdiv(N, 16) * (64 / 16);
        k_gemm_wmma<64><<<cdiv(waves, B / 32), B, 0, stream>>>(x, W1, h1, N);
    }
    k_fill<<<cdiv((long long)N * 64, B), B, 0, stream>>>(z1, 0.0f, (long long)N * 64);
    k_scatter<64><<<cdiv(E * 16, B), B, 0, stream>>>(src, dst, dinv, h1, z1, E);
    k_finish<64, true><<<cdiv((long long)N * 64, B), B, 0, stream>>>(z1, h1, dinv, b1, N);

    // ---- Layer 2: h2 = z1 @ W2 ; agg into d_out ; + dinv^2*h2 + b2 ----
    {
        const int waves = cdiv(N, 16) * (32 / 16);
        k_gemm_wmma<32><<<cdiv(waves, B / 32), B, 0, stream>>>(z1, W2, h2, N);
    }
    k_fill<<<cdiv((long long)N * 32, B), B, 0, stream>>>(out, 0.0f, (long long)N * 32);
    k_scatter<32><<<cdiv(E * 8, B), B, 0, stream>>>(src, dst, dinv, h2, out, E);
    k_finish<32, false><<<cdiv((long long)N * 32, B), B, 0, stream>>>(out, h2, dinv, b2, N);
}